// LSTM_14620068675655
// MI455X (gfx1250) — compile-verified
//
#include <hip/hip_runtime.h>
#include <cstdint>

// Problem dims (fixed by the reference)
#define BB       64
#define T_TOTAL  1024
#define DD       512
#define HH       512
#define G4       2048      // 4*HH gate columns, order [c|i|f|o]
#define HP       516       // LDS pitch for 512-wide rows (bank padding: 516%64==4)

typedef float v2f __attribute__((ext_vector_type(2)));
typedef float v8f __attribute__((ext_vector_type(8)));
typedef int   v4i __attribute__((ext_vector_type(4)));

// ---------------------------------------------------------------------------
// CDNA5 async global->LDS copies (ASYNCcnt). Parameter types confirmed by
// compiler diagnostics: (AS1 v4i*/int* src, AS3 dst, i32 offset, i32 cpol).
// ---------------------------------------------------------------------------
#define AS1 __attribute__((address_space(1)))
#define AS3 __attribute__((address_space(3)))

#if defined(__has_builtin)
#if __has_builtin(__builtin_amdgcn_global_load_async_to_lds_b128) && \
    __has_builtin(__builtin_amdgcn_global_load_async_to_lds_b32)
#define HAVE_ASYNC_LDS 1
#endif
#endif
#ifndef HAVE_ASYNC_LDS
#define HAVE_ASYNC_LDS 0
#endif

__device__ __forceinline__ void async_copy16(const float* g, float* l) {
#if HAVE_ASYNC_LDS
  __builtin_amdgcn_global_load_async_to_lds_b128((AS1 v4i*)g, (AS3 v4i*)l, 0,
                                                 0);
#else
  *(float4*)l = *(const float4*)g;
#endif
}

__device__ __forceinline__ void async_copy4(const float* g, float* l) {
#if HAVE_ASYNC_LDS
  __builtin_amdgcn_global_load_async_to_lds_b32((AS1 int*)g, (AS3 int*)l, 0,
                                                0);
#else
  *l = *g;
#endif
}

__device__ __forceinline__ void async_wait0() {
#if HAVE_ASYNC_LDS
#if __has_builtin(__builtin_amdgcn_s_wait_asynccnt)
  __builtin_amdgcn_s_wait_asynccnt(0);
#else
  asm volatile("s_wait_asynccnt 0x0" ::: "memory");
#endif
#endif
}

// V_WMMA_F32_16X16X4_F32 : D(16x16,f32) = A(16x4,f32) * B(4x16,f32) + C
__device__ __forceinline__ v8f wmma_f32(v2f a, v2f b, v8f c) {
  return __builtin_amdgcn_wmma_f32_16x16x4_f32(false, a, false, b, (short)0, c,
                                               false, false);
}

__device__ __forceinline__ float sigmoidf_(float x) {
  return 1.0f / (1.0f + __expf(-x));
}

// ---------------------------------------------------------------------------
// Phase 1: xproj[b*TC+tt, gc] = sum_k x[b, t0+tt, k] * W[k, gc] + bias[gc]
// 64x128 tile per 256-thread block; K-chunks double-buffered via async-to-LDS.
// ---------------------------------------------------------------------------
__global__ __launch_bounds__(256) void lstm_xproj_kernel(
    const float* __restrict__ x,
    const float* __restrict__ Wc, const float* __restrict__ Wi,
    const float* __restrict__ Wf, const float* __restrict__ Wo,
    const float* __restrict__ bc, const float* __restrict__ bi,
    const float* __restrict__ bf, const float* __restrict__ bo,
    float* __restrict__ xbuf, int t0, int TC) {
  __shared__ float xs[2][64 * 36];    // x tiles  [64 rows][32 k], pitch 36
  __shared__ float ws[2][128 * 36];   // W tiles, transposed [128 n][32 k]

  const int rr0 = blockIdx.y * 64;            // row in [0, BB*TC)
  const int b   = rr0 / TC;                   // batch index (TC % 64 == 0)
  const int tt0 = rr0 % TC;
  const float* xrow = x + ((size_t)b * T_TOTAL + t0 + tt0) * DD;

  const int gc0 = blockIdx.x * 128;           // gate-column base
  const int g   = gc0 >> 9;
  const int wc0 = gc0 & 511;
  const float* Wg = (g == 0) ? Wc : (g == 1) ? Wi : (g == 2) ? Wf : Wo;
  const float* bg = (g == 0) ? bc : (g == 1) ? bi : (g == 2) ? bf : bo;

  const int tid  = threadIdx.x;
  const int wv   = tid >> 5;        // wave 0..7 -> N subtile
  const int lane = tid & 31;
  const int lm   = lane & 15;
  const int lh   = lane >> 4;

  auto issue_chunk = [&](int kb, float* xsb, float* wsb) {
    // x tile (64x32) as b128 async
    for (int idx = tid; idx < 512; idx += 256) {
      int r = idx >> 3, k4 = idx & 7;
      async_copy16(xrow + (size_t)r * DD + kb + k4 * 4, &xsb[r * 36 + k4 * 4]);
    }
    // W tile transposed (128n x 32k) as b32 async scatter (n-contiguous reads)
    for (int idx = tid; idx < 4096; idx += 256) {
      int n = idx & 127, k = idx >> 7;
      async_copy4(Wg + (size_t)(kb + k) * HH + wc0 + n, &wsb[n * 36 + k]);
    }
  };

  v8f acc[4] = {};                  // 4 M-subtiles x (16x16)
  issue_chunk(0, xs[0], ws[0]);

  for (int cb = 0; cb < DD / 32; ++cb) {
    async_wait0();
    __syncthreads();                // current tiles fully in LDS; prev compute done
    if (cb + 1 < DD / 32)
      issue_chunk((cb + 1) * 32, xs[(cb + 1) & 1], ws[(cb + 1) & 1]);

    const float* bp = ws[cb & 1] + (wv * 16 + lm) * 36 + 2 * lh;
    const float* ap = xs[cb & 1] + lm * 36 + 2 * lh;
#pragma unroll
    for (int kk = 0; kk < 32; kk += 4) {
      v2f bfrag = *(const v2f*)(bp + kk);
#pragma unroll
      for (int mi = 0; mi < 4; ++mi) {
        v2f afrag = *(const v2f*)(ap + mi * 16 * 36 + kk);
        acc[mi] = wmma_f32(afrag, bfrag, acc[mi]);
      }
    }
  }

  const float bias = bg[wc0 + wv * 16 + lm];
#pragma unroll
  for (int mi = 0; mi < 4; ++mi) {
#pragma unroll
    for (int r = 0; r < 8; ++r) {
      int m = mi * 16 + r + 8 * lh;
      xbuf[((size_t)(rr0 + m)) * G4 + gc0 + wv * 16 + lm] = acc[mi][r] + bias;
    }
  }
}

// ---------------------------------------------------------------------------
// Device-wide barrier: monotonically increasing counter, release/acquire.
// ---------------------------------------------------------------------------
__device__ __forceinline__ void grid_sync(unsigned* counter, unsigned target) {
  __syncthreads();
  if (threadIdx.x == 0) {
    __hip_atomic_fetch_add(counter, 1u, __ATOMIC_RELEASE,
                           __HIP_MEMORY_SCOPE_AGENT);
    while (__hip_atomic_load(counter, __ATOMIC_ACQUIRE,
                             __HIP_MEMORY_SCOPE_AGENT) < target) {
      __builtin_amdgcn_s_sleep(1);
    }
  }
  __syncthreads();
}

// ---------------------------------------------------------------------------
// Phase 2: persistent scan over TC timesteps. 32 WGs; WG wg owns h columns
// [wg*16, wg*16+16). LDS: full h (64x512), U slice (transposed [gc][k]),
// gate activations, c state, and a double-buffered xproj prefetch tile.
// ---------------------------------------------------------------------------
__global__ __launch_bounds__(256) void lstm_scan_kernel(
    const float* __restrict__ xbuf,                 // [BB*TC, G4]
    const float* __restrict__ Uc, const float* __restrict__ Ui,
    const float* __restrict__ Uf, const float* __restrict__ Uo,
    float* __restrict__ hbuf,                       // [2][BB*HH] double buffer
    float* __restrict__ cbuf,                       // [BB*HH]
    float* __restrict__ out,                        // [BB, T_TOTAL, HH]
    unsigned* __restrict__ counter, int t0, int TC) {
  extern __shared__ float smem[];
  float* h_lds  = smem;                   // [64][HP]
  float* u_lds  = h_lds + 64 * HP;        // [64 gc][HP], [g*16+n][k]
  float* g_lds  = u_lds + 64 * HP;        // [4*64][17] gate activations
  float* c_lds  = g_lds + 4 * 64 * 17;    // [64][16]
  float* xp_lds = c_lds + 64 * 16;        // [2][4][64][16] xproj prefetch

  const int col0 = blockIdx.x * 16;       // owned h-column base
  const int tid  = threadIdx.x;
  const int wv   = tid >> 5;              // 8 waves
  const int lane = tid & 31;
  const int lm   = lane & 15;
  const int lh   = lane >> 4;
  const int g    = wv & 3;                // gate for this wave
  const int m0   = (wv >> 2) * 32;        // rows m0..m0+31

  auto issue_xp = [&](int tt, float* dst) {   // 4 gates x 64 m x 16 n
    for (int idx = tid; idx < 1024; idx += 256) {
      int q = idx & 3, m = (idx >> 2) & 63, gg = idx >> 8;
      async_copy16(&xbuf[((size_t)m * TC + tt) * G4 + gg * 512 + col0 + q * 4],
                   &dst[(gg * 64 + m) * 16 + q * 4]);
    }
  };

  // Stage this WG's U slice into LDS once (reused TC times)
  const float* Us[4] = {Uc, Ui, Uf, Uo};
  for (int gg = 0; gg < 4; ++gg) {
    const float* U = Us[gg];
    for (int idx = tid; idx < 16 * HH; idx += 256) {
      int n = idx & 15, k = idx >> 4;
      u_lds[(gg * 16 + n) * HP + k] = U[(size_t)k * HH + col0 + n];
    }
  }
  for (int idx = tid; idx < 64 * 16; idx += 256)
    c_lds[idx] = cbuf[(size_t)(idx >> 4) * HH + col0 + (idx & 15)];
  issue_xp(0, xp_lds);                    // prefetch xproj for tt = 0
  __syncthreads();

  for (int tt = 0; tt < TC; ++tt) {
    const int t = t0 + tt;
    const float* hcur = hbuf + (size_t)(t & 1) * (BB * HH);
    float* hnext      = hbuf + (size_t)((t + 1) & 1) * (BB * HH);

    // Broadcast full h into LDS via async b128 (L2-resident)
    for (int idx = tid; idx < 64 * (HH / 4); idx += 256) {
      int m = idx >> 7, k4 = idx & 127;
      async_copy16(hcur + (size_t)m * HH + k4 * 4, &h_lds[m * HP + k4 * 4]);
    }
    async_wait0();                        // h ready + this step's xp ready
    __syncthreads();

    // Prefetch next step's xproj tile; hides under the WMMA chain below.
    if (tt + 1 < TC) issue_xp(tt + 1, xp_lds + ((tt + 1) & 1) * 4096);

    // GEMM: rows [m0, m0+32) x 16 cols of gate g
    const float* bp  = u_lds + (g * 16 + lm) * HP + 2 * lh;
    const float* ap0 = h_lds + (m0 + lm) * HP + 2 * lh;
    const float* ap1 = ap0 + 16 * HP;
    v8f acc0 = {}, acc1 = {};
#pragma unroll 8
    for (int k = 0; k < HH; k += 4) {
      v2f bfrag = *(const v2f*)(bp + k);
      v2f a0    = *(const v2f*)(ap0 + k);
      v2f a1    = *(const v2f*)(ap1 + k);
      acc0 = wmma_f32(a0, bfrag, acc0);
      acc1 = wmma_f32(a1, bfrag, acc1);
    }

    // Epilogue: add xproj (from LDS prefetch), apply gate nonlinearity
    const float* xp = xp_lds + (tt & 1) * 4096;
#pragma unroll
    for (int mi = 0; mi < 2; ++mi) {
      v8f acc = mi ? acc1 : acc0;
#pragma unroll
      for (int r = 0; r < 8; ++r) {
        int m = m0 + mi * 16 + r + 8 * lh;   // batch row
        float pre = acc[r] + xp[(g * 64 + m) * 16 + lm];
        float act = (g == 0) ? tanhf(pre) : sigmoidf_(pre);
        g_lds[(g * 64 + m) * 17 + lm] = act;
      }
    }
    __syncthreads();

    // Elementwise LSTM cell update for owned [64 x 16] block
    for (int idx = tid; idx < 64 * 16; idx += 256) {
      int m = idx >> 4, n = idx & 15;
      float a = g_lds[(0 * 64 + m) * 17 + n];
      float i = g_lds[(1 * 64 + m) * 17 + n];
      float f = g_lds[(2 * 64 + m) * 17 + n];
      float o = g_lds[(3 * 64 + m) * 17 + n];
      float c  = c_lds[idx];
      float cn = i * a + f * c;
      float hn = o * tanhf(cn);
      c_lds[idx] = cn;
      hnext[(size_t)m * HH + col0 + n] = hn;
      out[((size_t)m * T_TOTAL + t) * HH + col0 + n] = hn;
    }

    grid_sync(counter, (unsigned)(gridDim.x * (tt + 1)));
  }

  // Persist c state for next chunk
  for (int idx = tid; idx < 64 * 16; idx += 256)
    cbuf[(size_t)(idx >> 4) * HH + col0 + (idx & 15)] = c_lds[idx];
}

// ---------------------------------------------------------------------------
// Host launch: chunk T into TC-step pieces; per chunk: xproj GEMM, then scan.
// ---------------------------------------------------------------------------
extern "C" void kernel_launch(void* const* d_in, const int* in_sizes, int n_in,
                              void* d_out, int out_size, void* d_ws,
                              size_t ws_size, hipStream_t stream) {
  (void)in_sizes; (void)n_in; (void)out_size;
  const float* x  = (const float*)d_in[0];
  const float* Wc = (const float*)d_in[1];
  const float* Wi = (const float*)d_in[2];
  const float* Wf = (const float*)d_in[3];
  const float* Wo = (const float*)d_in[4];
  const float* Uc = (const float*)d_in[5];
  const float* Ui = (const float*)d_in[6];
  const float* Uf = (const float*)d_in[7];
  const float* Uo = (const float*)d_in[8];
  const float* bc = (const float*)d_in[9];
  const float* bi = (const float*)d_in[10];
  const float* bf = (const float*)d_in[11];
  const float* bo = (const float*)d_in[12];
  float* out = (float*)d_out;

  // Chunk size: xproj scratch = BB*TC*G4 floats. Prefer 128 (64 MB).
  int TC = 128;
  {
    size_t need = (size_t)BB * TC * G4 * 4 + (size_t)3 * BB * HH * 4 + 256;
    if (ws_size < need) TC = 64;   // 34 MB fallback
  }

  float*    xbuf    = (float*)d_ws;
  float*    hbuf    = xbuf + (size_t)BB * TC * G4;   // [2][BB*HH]
  float*    cbuf    = hbuf + (size_t)2 * BB * HH;    // [BB*HH]
  unsigned* counter = (unsigned*)(cbuf + (size_t)BB * HH);

  // Zero h double-buffer, c state, and barrier counter (graph-capturable).
  (void)hipMemsetAsync(hbuf, 0, (size_t)3 * BB * HH * sizeof(float) + 64,
                       stream);

  const size_t scan_lds =
      (size_t)(2 * 64 * HP + 4 * 64 * 17 + 64 * 16 + 2 * 4096) *
      sizeof(float);  // ~311 KB

  for (int t0 = 0; t0 < T_TOTAL; t0 += TC) {
    (void)hipMemsetAsync(counter, 0, sizeof(unsigned), stream);
    dim3 g1(G4 / 128, (BB * TC) / 64);
    lstm_xproj_kernel<<<g1, 256, 0, stream>>>(x, Wc, Wi, Wf, Wo,
                                              bc, bi, bf, bo, xbuf, t0, TC);
    lstm_scan_kernel<<<HH / 16, 256, scan_lds, stream>>>(
        xbuf, Uc, Ui, Uf, Uo, hbuf, cbuf, out, counter, t0, TC);
  }
}